// LogGD_7773890806276
// MI455X (gfx1250) — compile-verified
//
#include <hip/hip_runtime.h>
#include <math.h>

#define N_NODES 4096
#define N_EDGES 8192
#define IN_F 256
#define OUT_F 256

typedef __attribute__((ext_vector_type(2))) float v2f;
typedef __attribute__((ext_vector_type(8))) float v8f;

// CDNA5 f32 WMMA: D(16x16) = A(16x4) * B(4x16) + C   (wave32, VOP3P)
__device__ __forceinline__ v8f wmma_f32(v2f a, v2f b, v8f c) {
  return __builtin_amdgcn_wmma_f32_16x16x4_f32(false, a, false, b, (short)0, c,
                                               false, false);
}

// Low 32 bits of a generic pointer to __shared__ = wave-relative LDS offset
// (ISA 10.2).
__device__ __forceinline__ unsigned int lds_off(const void* p) {
  return (unsigned int)(unsigned long long)p;
}

// GLOBAL_LOAD_ASYNC_TO_LDS_B128: 16B/lane global -> LDS, tracked by ASYNCcnt.
__device__ __forceinline__ void async_ld_b128(unsigned int dst_lds,
                                              const void* src_global) {
  asm volatile("global_load_async_to_lds_b128 %0, %1, off"
               :: "v"(dst_lds), "v"(src_global)
               : "memory");
}

__device__ __forceinline__ void wait_async0() {
  asm volatile("s_wait_asynccnt 0x0" ::: "memory");
}

__device__ __forceinline__ void atomicMaxFloat(float* addr, float val) {
  // sign-aware float max via integer atomics (addr initialized to -inf)
  if (val >= 0.0f) atomicMax((int*)addr, __float_as_int(val));
  else             atomicMin((unsigned int*)addr, __float_as_uint(val));
}

__device__ __forceinline__ float gelu_exact(float x) {
  return 0.5f * x * (1.0f + erff(x * 0.70710678118654752f));
}

// ---------------------------------------------------------------- init
__global__ void k_init(int* in_deg, int* out_deg, float* zsum, float* zmax) {
  int i = blockIdx.x * 256 + threadIdx.x;
  if (i < N_NODES) { in_deg[i] = 0; out_deg[i] = 0; }
  if (i < OUT_F)   { zsum[i] = 0.0f; zmax[i] = -INFINITY; }
}

// ---------------------------------------------------------------- degrees
__global__ void k_deg(const int* ei, int* in_deg, int* out_deg) {
  int e = blockIdx.x * 256 + threadIdx.x;
  if (e < N_EDGES) {
    atomicAdd(&in_deg[ei[e]], 1);            // src
    atomicAdd(&out_deg[ei[N_EDGES + e]], 1); // dst
  }
}

// ------------------------------------------------- x0 = x + z_in[d] + z_out[d]
__global__ void k_x0(const float* x, const float* z_in, const float* z_out,
                     const int* in_deg, const int* out_deg, float* x0) {
  int idx = blockIdx.x * 256 + threadIdx.x; // N_NODES*IN_F
  int node = idx >> 8, f = idx & 255;
  int di = min(in_deg[node], IN_F - 1);   // JAX gather clamps OOB
  int dd = min(out_deg[node], IN_F - 1);
  x0[idx] = x[idx] + z_in[di * OUT_F + f] + z_out[dd * OUT_F + f];
}

// -------------------------------------------- Q0 = w*(xs Wq^T + b), K, Vd=w*(V+b)+D
__global__ __launch_bounds__(256) void k_qkv(
    const float* __restrict__ x0, const int* __restrict__ ei,
    const float* __restrict__ ew,
    const float* __restrict__ Wq_w, const float* __restrict__ Wq_b,
    const float* __restrict__ Wk_w, const float* __restrict__ Wk_b,
    const float* __restrict__ Wv_w, const float* __restrict__ Wv_b,
    const float* __restrict__ D,
    float* __restrict__ Q0, float* __restrict__ Km, float* __restrict__ Vd) {
  __shared__ float Xs[16 * 256];
  __shared__ float Xd[16 * 256];
  int tid = threadIdx.x;
  int e0 = blockIdx.x * 16;
  // async gather of 16 src rows + 16 dst rows (1KB each) straight into LDS
  for (int idx = tid; idx < 16 * 64; idx += 256) {   // 16B granules
    int r = idx >> 6, g = (idx & 63) * 4;
    async_ld_b128(lds_off(&Xs[r * 256 + g]), &x0[ei[e0 + r] * 256 + g]);
    async_ld_b128(lds_off(&Xd[r * 256 + g]), &x0[ei[N_EDGES + e0 + r] * 256 + g]);
  }
  wait_async0();
  __syncthreads();

  int lane = tid & 31, wave = tid >> 5;
  int n = lane & 15, half = lane >> 4;
  // 48 (matrix, out-tile) pairs over 8 waves -> 6 tiles each, uniform per wave
  for (int p = wave; p < 48; p += 8) {
    int mat = p >> 4;    // 0=Q 1=K 2=V
    int tcol = p & 15;   // output-feature tile
    const float* W = (mat == 0) ? Wq_w : (mat == 1) ? Wk_w : Wv_w;
    const float* B = (mat == 0) ? Wq_b : (mat == 1) ? Wk_b : Wv_b;
    const float* X = (mat == 0) ? Xs : Xd;
    int o = tcol * 16 + n;
    v8f c = {0.f, 0.f, 0.f, 0.f, 0.f, 0.f, 0.f, 0.f};
    for (int kk = 0; kk < 64; ++kk) {
      int kb = kk * 4 + 2 * half;
      v2f a, b;
      a.x = X[n * 256 + kb];        // A[m=lane%16][k]
      a.y = X[n * 256 + kb + 1];
      b.x = W[o * 256 + kb];        // B[k][n] = W[o][k]
      b.y = W[o * 256 + kb + 1];
      c = wmma_f32(a, b, c);
    }
    float bias = B[o];
    float* OutP = (mat == 0) ? Q0 : (mat == 1) ? Km : Vd;
    float addb = (mat == 2) ? D[o] : 0.f;
#pragma unroll
    for (int v = 0; v < 8; ++v) {
      int m = v + 8 * half;
      int e = e0 + m;
      float mult = (mat == 1) ? 1.0f : ew[e];
      OutP[e * 256 + o] = mult * (c[v] + bias) + addb;
    }
  }
}

// ------------------------------------- cb[j] = (Q0[j]+K[j]) . D  (column bias)
__global__ void k_cbias(const float* __restrict__ Q0, const float* __restrict__ Km,
                        const float* __restrict__ D, float* __restrict__ cb) {
  int gid = blockIdx.x * 256 + threadIdx.x;
  int e = gid >> 5, lane = gid & 31;
  if (e >= N_EDGES) return;
  float acc = 0.f;
  for (int o = lane; o < OUT_F; o += 32)
    acc += (Q0[e * 256 + o] + Km[e * 256 + o]) * D[o];
  for (int m = 16; m; m >>= 1) acc += __shfl_xor(acc, m, 32);
  if (lane == 0) cb[e] = acc;
}

// --------------------- fused flash attention + column sum/max reduction of z
__global__ __launch_bounds__(256) void k_attn(
    const float* __restrict__ Q0, const float* __restrict__ Km,
    const float* __restrict__ Vd, const float* __restrict__ cb,
    float* __restrict__ zsum, float* __restrict__ zmax) {
  __shared__ float Qs[16 * 256];   // 16 query rows
  __shared__ float Pl[16 * 128];   // S then P tile for current 128-col chunk
  __shared__ float mrow[16], lrow[16], rfac[16];
  int tid = threadIdx.x;
  int i0 = blockIdx.x * 16;
  // async-stage Q tile (16KB, contiguous) into LDS
  for (int idx = tid; idx < 16 * 64; idx += 256)
    async_ld_b128(lds_off(&Qs[idx * 4]), &Q0[i0 * 256 + idx * 4]);
  if (tid < 16) { mrow[tid] = -INFINITY; lrow[tid] = 0.f; }
  wait_async0();
  __syncthreads();

  int lane = tid & 31, wave = tid >> 5;
  int n = lane & 15, half = lane >> 4;
  int fbase = wave * 32;  // this wave owns output features [fbase, fbase+32)
  v8f acc0 = {0.f, 0.f, 0.f, 0.f, 0.f, 0.f, 0.f, 0.f};
  v8f acc1 = {0.f, 0.f, 0.f, 0.f, 0.f, 0.f, 0.f, 0.f};
  const float scale = 0.0625f;  // 1/sqrt(256)

  for (int jc = 0; jc < N_EDGES; jc += 128) {
    int j0 = jc + wave * 16;
    // prefetch this wave's next K chunk (16 rows = 16KB contiguous)
    if (jc + 128 < N_EDGES) {
      const float* nk = &Km[(j0 + 128) * 256];
#pragma unroll
      for (int pp = 0; pp < 4; ++pp)
        __builtin_prefetch(nk + (pp * 32 + lane) * 32, 0, 3);  // 128B/lane
    }
    // --- S tile: each wave computes 16 rows x its 16 columns
    v8f s = {0.f, 0.f, 0.f, 0.f, 0.f, 0.f, 0.f, 0.f};
    for (int kk = 0; kk < 64; ++kk) {
      int kb = kk * 4 + 2 * half;
      v2f a, b;
      a.x = Qs[n * 256 + kb];
      a.y = Qs[n * 256 + kb + 1];
      b.x = Km[(j0 + n) * 256 + kb];      // B[k][n] = K[j0+n][k]
      b.y = Km[(j0 + n) * 256 + kb + 1];
      s = wmma_f32(a, b, s);
    }
    float cbn = cb[j0 + n];
#pragma unroll
    for (int v = 0; v < 8; ++v) {
      int m = v + 8 * half;                     // C layout: row = v + 8*half
      Pl[m * 128 + wave * 16 + n] = s[v] * scale + cbn;
    }
    __syncthreads();
    // --- online-softmax stats (one thread per row)
    if (tid < 16) {
      float tmax = -INFINITY;
      for (int j = 0; j < 128; ++j) tmax = fmaxf(tmax, Pl[tid * 128 + j]);
      float newm = fmaxf(mrow[tid], tmax);
      float r = __expf(mrow[tid] - newm);
      rfac[tid] = r;
      mrow[tid] = newm;
      lrow[tid] *= r;
    }
    __syncthreads();
    // --- exponentiate tile in place
    for (int idx = tid; idx < 16 * 128; idx += 256) {
      int r = idx >> 7;
      Pl[idx] = __expf(Pl[idx] - mrow[r]);
    }
    __syncthreads();
    if (tid < 16) {
      float srow = 0.f;
      for (int j = 0; j < 128; ++j) srow += Pl[tid * 128 + j];
      lrow[tid] += srow;
    }
    // --- rescale running accumulator and do z += P(16x128) @ Vd(128x32)
#pragma unroll
    for (int v = 0; v < 8; ++v) {
      float rf = rfac[v + 8 * half];
      acc0[v] *= rf;
      acc1[v] *= rf;
    }
    for (int kk = 0; kk < 32; ++kk) {
      int kb = kk * 4 + 2 * half;
      v2f a, b0, b1;
      a.x = Pl[n * 128 + kb];
      a.y = Pl[n * 128 + kb + 1];
      b0.x = Vd[(jc + kb) * 256 + fbase + n];
      b0.y = Vd[(jc + kb + 1) * 256 + fbase + n];
      b1.x = Vd[(jc + kb) * 256 + fbase + 16 + n];
      b1.y = Vd[(jc + kb + 1) * 256 + fbase + 16 + n];
      acc0 = wmma_f32(a, b0, acc0);
      acc1 = wmma_f32(a, b1, acc1);
    }
    __syncthreads();  // protect Pl / rfac before next chunk
  }

  // --- normalize, reduce this block's 16 rows into column sum / max
  v8f* accs[2] = {&acc0, &acc1};
#pragma unroll
  for (int t = 0; t < 2; ++t) {
    v8f a = *accs[t];
    float csum = 0.f, cmax = -INFINITY;
#pragma unroll
    for (int v = 0; v < 8; ++v) {
      float z = a[v] / lrow[v + 8 * half];
      csum += z;
      cmax = fmaxf(cmax, z);
    }
    csum += __shfl_xor(csum, 16, 32);                 // combine halves (rows 0-7 / 8-15)
    cmax = fmaxf(cmax, __shfl_xor(cmax, 16, 32));
    if (half == 0) {
      int f = fbase + t * 16 + n;
      atomicAdd(&zsum[f], csum);
      atomicMaxFloat(&zmax[f], cmax);
    }
  }
}

// ------------------------------------------------ LayerNorm over hg (512)
__global__ void k_ln(const float* zsum, const float* zmax, const float* ln_g,
                     const float* ln_b, float* hgn) {
  __shared__ float sh[512];
  __shared__ float red[512];
  int t = threadIdx.x;
  float v = (t < 256) ? zsum[t] : zmax[t - 256];
  sh[t] = v;
  red[t] = v;
  __syncthreads();
  for (int s = 256; s > 0; s >>= 1) {
    if (t < s) red[t] += red[t + s];
    __syncthreads();
  }
  float mu = red[0] * (1.0f / 512.0f);
  __syncthreads();
  float d = sh[t] - mu;
  red[t] = d * d;
  __syncthreads();
  for (int s = 256; s > 0; s >>= 1) {
    if (t < s) red[t] += red[t + s];
    __syncthreads();
  }
  float var = red[0] * (1.0f / 512.0f);
  hgn[t] = d * rsqrtf(var + 1e-5f) * ln_g[t] + ln_b[t];
}

// ------------------------------------------------ small dense layers
__global__ void k_fc(const float* __restrict__ W, const float* __restrict__ b,
                     const float* __restrict__ in, float* __restrict__ out,
                     int in_dim, int do_gelu) {
  int o = threadIdx.x;  // 256 threads
  float acc = b[o];
  for (int i = 0; i < in_dim; ++i) acc += W[o * in_dim + i] * in[i];
  out[o] = do_gelu ? gelu_exact(acc) : acc;
}

__global__ void k_fc3(const float* __restrict__ W, const float* __restrict__ b,
                      const float* __restrict__ in, float* __restrict__ out) {
  int w = threadIdx.x >> 5, lane = threadIdx.x & 31;  // 64 threads = 2 waves
  float acc = 0.f;
  for (int i = lane; i < OUT_F; i += 32) acc += W[w * OUT_F + i] * in[i];
  for (int m = 16; m; m >>= 1) acc += __shfl_xor(acc, m, 32);
  if (lane == 0) out[w] = b[w] + acc;
}

// ---------------------------------------------------------------- launch
extern "C" void kernel_launch(void* const* d_in, const int* in_sizes, int n_in,
                              void* d_out, int out_size, void* d_ws, size_t ws_size,
                              hipStream_t stream) {
  const float* x     = (const float*)d_in[0];
  const int*   ei    = (const int*)d_in[1];
  const float* ew    = (const float*)d_in[2];
  const float* Wq_w  = (const float*)d_in[3];
  const float* Wq_b  = (const float*)d_in[4];
  const float* Wk_w  = (const float*)d_in[5];
  const float* Wk_b  = (const float*)d_in[6];
  const float* Wv_w  = (const float*)d_in[7];
  const float* Wv_b  = (const float*)d_in[8];
  const float* z_in  = (const float*)d_in[9];
  const float* z_out = (const float*)d_in[10];
  const float* D     = (const float*)d_in[11];
  const float* ln_g  = (const float*)d_in[12];
  const float* ln_b  = (const float*)d_in[13];
  const float* fc1_w = (const float*)d_in[14];
  const float* fc1_b = (const float*)d_in[15];
  const float* fc2_w = (const float*)d_in[16];
  const float* fc2_b = (const float*)d_in[17];
  const float* fc3_w = (const float*)d_in[18];
  const float* fc3_b = (const float*)d_in[19];
  float* out = (float*)d_out;

  // workspace layout (~29.5 MB total)
  char* w = (char*)d_ws;
  int* in_deg   = (int*)w;   w += (size_t)N_NODES * 4;
  int* out_deg  = (int*)w;   w += (size_t)N_NODES * 4;
  float* x0     = (float*)w; w += (size_t)N_NODES * IN_F * 4;
  float* Q0     = (float*)w; w += (size_t)N_EDGES * OUT_F * 4;
  float* Km     = (float*)w; w += (size_t)N_EDGES * OUT_F * 4;
  float* Vd     = (float*)w; w += (size_t)N_EDGES * OUT_F * 4;
  float* cbv    = (float*)w; w += (size_t)N_EDGES * 4;
  float* zsum   = (float*)w; w += OUT_F * 4;
  float* zmax   = (float*)w; w += OUT_F * 4;
  float* hgn    = (float*)w; w += 512 * 4;
  float* h1     = (float*)w; w += OUT_F * 4;
  float* h2     = (float*)w; w += OUT_F * 4;

  k_init <<<N_NODES / 256, 256, 0, stream>>>(in_deg, out_deg, zsum, zmax);
  k_deg  <<<N_EDGES / 256, 256, 0, stream>>>(ei, in_deg, out_deg);
  k_x0   <<<N_NODES * IN_F / 256, 256, 0, stream>>>(x, z_in, z_out, in_deg, out_deg, x0);
  k_qkv  <<<N_EDGES / 16, 256, 0, stream>>>(x0, ei, ew, Wq_w, Wq_b, Wk_w, Wk_b,
                                            Wv_w, Wv_b, D, Q0, Km, Vd);
  k_cbias<<<N_EDGES * 32 / 256, 256, 0, stream>>>(Q0, Km, D, cbv);
  k_attn <<<N_EDGES / 16, 256, 0, stream>>>(Q0, Km, Vd, cbv, zsum, zmax);
  k_ln   <<<1, 512, 0, stream>>>(zsum, zmax, ln_g, ln_b, hgn);
  k_fc   <<<1, 256, 0, stream>>>(fc1_w, fc1_b, hgn, h1, 2 * OUT_F, 1);
  k_fc   <<<1, 256, 0, stream>>>(fc2_w, fc2_b, h1, h2, OUT_F, 1);
  k_fc3  <<<1, 64, 0, stream>>>(fc3_w, fc3_b, h2, out);
}